// RanGINJK_virtualnode_13039520711153
// MI455X (gfx1250) — compile-verified
//
#include <hip/hip_runtime.h>

typedef float v2f __attribute__((ext_vector_type(2)));
typedef float v8f __attribute__((ext_vector_type(8)));

constexpr int HD = 128;   // hidden size (fixed by problem)

// ---------------- one-time init: degree=0, vn broadcast, vt=0, graph starts --
__global__ void init_all(float* __restrict__ deg, float* __restrict__ vn,
                         float* __restrict__ vt, const float* __restrict__ vn_emb,
                         const int* __restrict__ batch, int* __restrict__ starts,
                         int N, int G)
{
    int i = blockIdx.x * blockDim.x + threadIdx.x;
    if (i < N) deg[i] = 0.f;
    if (i < G * HD) { vn[i] = vn_emb[i & (HD - 1)]; vt[i] = 0.f; }
    if (i <= G) {
        int lo = 0, hi = N;                       // lower_bound(batch, i)
        while (lo < hi) { int mid = (lo + hi) >> 1; if (batch[mid] < i) lo = mid + 1; else hi = mid; }
        starts[i] = lo;
    }
}

__global__ void deg_accum(const int* __restrict__ tgt, float* __restrict__ deg, int E)
{
    int e = blockIdx.x * blockDim.x + threadIdx.x;
    if (e < E) atomicAdd(&deg[tgt[e]], 1.f);
}

// ---------------- h = hin + vn[batch]; agg = 0 -------------------------------
__global__ void add_vn_zero_agg(const float* __restrict__ hin, long ldin,
                                const float* __restrict__ vn, const int* __restrict__ batch,
                                float* __restrict__ h, float* __restrict__ agg, int N)
{
    long i = (long)blockIdx.x * blockDim.x + threadIdx.x;
    if (i < (long)N * HD) {
        int n = (int)(i >> 7);
        int c = (int)(i & (HD - 1));
        h[i] = hin[(long)n * ldin + c] + vn[(batch[n] << 7) + c];
        agg[i] = 0.f;
    }
}

// ---------------- per-edge: msg = relu(h[src] + ea*eW + eb); agg[tgt] += msg -
__global__ void edge_scatter(const float* __restrict__ h, const float* __restrict__ ea,
                             const int* __restrict__ src, const int* __restrict__ tgt,
                             const float* __restrict__ eW, const float* __restrict__ eb,
                             float* __restrict__ agg)
{
    int e = blockIdx.x;
    int c = threadIdx.x;                      // 128 threads = one row of channels
    float a = ea[e];
    int s = src[e], t = tgt[e];
    float v = h[((long)s << 7) + c] + a * eW[c] + eb[c];
    v = fmaxf(v, 0.f);
    atomicAdd(&agg[((long)t << 7) + c], v);
}

// ---------------- z = (1+eps)*h + agg/max(deg,1) -----------------------------
__global__ void make_z(const float* __restrict__ h, const float* __restrict__ agg,
                       const float* __restrict__ deg, const float* __restrict__ eps_arr,
                       int layer, float* __restrict__ z, int N)
{
    long i = (long)blockIdx.x * blockDim.x + threadIdx.x;
    if (i < (long)N * HD) {
        int n = (int)(i >> 7);
        float e1 = 1.f + eps_arr[layer];
        z[i] = e1 * h[i] + agg[i] / fmaxf(deg[n], 1.f);
    }
}

// ---------------- Y[r, 0:128] = (relu?)(A[r,0:128] @ W + bias), WMMA f32 -----
// 256 threads = 8 waves; wave w owns rows [blk*128 + 16w, +16); cols 0..127.
// K loop in steps of 4 using V_WMMA_F32_16X16X4_F32.
// Weights staged in LDS K-pair-packed: sBp[k/2][col] = {W[k][col], W[k+1][col]}
// so each B fragment is a single aligned ds_load_b64 (no register shuffles).
__global__ void __launch_bounds__(256)
gemm_wmma_f32(const float* __restrict__ A, const float* __restrict__ W,
              const float* __restrict__ bias, float* __restrict__ Y, long ldY,
              int N, int relu)
{
    __shared__ float sBp[HD * HD];            // 64 KB packed weight tile
    int tid = threadIdx.x;
    // Load W row-major (coalesced float4) and scatter into pair-packed layout.
    for (int idx = tid * 4; idx < HD * HD; idx += 256 * 4) {
        int k   = idx >> 7;                   // W row (K index)
        int col = idx & (HD - 1);             // W col (N index), multiple of 4
        float4 w = *(const float4*)&W[idx];
        int base = (k >> 1) * (2 * HD) + col * 2 + (k & 1);
        sBp[base + 0] = w.x;
        sBp[base + 2] = w.y;
        sBp[base + 4] = w.z;
        sBp[base + 6] = w.w;
    }
    __syncthreads();

    int wave  = tid >> 5;
    int lane  = tid & 31;
    int lhalf = lane >> 4;                    // 0: lanes 0-15, 1: lanes 16-31
    int l15   = lane & 15;
    int rowBase = blockIdx.x * 128 + wave * 16;
    int arow    = rowBase + l15;
    const float* Ar = A + (long)((arow < N) ? arow : (N - 1)) * HD;  // clamp OOB rows

    const v8f vzero = {0.f, 0.f, 0.f, 0.f, 0.f, 0.f, 0.f, 0.f};
    v8f acc[8];
#pragma unroll
    for (int n = 0; n < 8; ++n) acc[n] = vzero;

    for (int kt = 0; kt < HD; kt += 4) {
        // A fragment 16x4: lane half selects K pair {kt,kt+1} vs {kt+2,kt+3}
        v2f a = *(const v2f*)&Ar[kt + 2 * lhalf];
        int k2row = ((kt >> 1) + lhalf) * (2 * HD);
#pragma unroll
        for (int n = 0; n < 8; ++n) {
            v2f b = *(const v2f*)&sBp[k2row + (n * 16 + l15) * 2];
            acc[n] = __builtin_amdgcn_wmma_f32_16x16x4_f32(
                false, a, false, b, (short)0, acc[n], false, false);
        }
    }

#pragma unroll
    for (int n = 0; n < 8; ++n) {
        int col = n * 16 + l15;
        float bv = bias[col];
#pragma unroll
        for (int j = 0; j < 8; ++j) {
            int r = rowBase + j + 8 * lhalf;  // VGPR j -> M=j (lanes 0-15) / j+8
            if (r < N) {
                float v = acc[n][j] + bv;
                if (relu) v = fmaxf(v, 0.f);
                Y[(long)r * ldY + col] = v;
            }
        }
    }
}

// ---------------- vt[g] += sum over nodes of graph g (chunked partials) ------
__global__ void vt_accum(const float* __restrict__ hsrc, long ld,
                         const int* __restrict__ starts, float* __restrict__ vt)
{
    int g = blockIdx.x, s = blockIdx.y, c = threadIdx.x;
    int st = starts[g], en = starts[g + 1];
    int S = gridDim.y;
    int cnt = en - st;
    int chunk = (cnt + S - 1) / S;
    int b = st + s * chunk;
    int e = b + chunk; if (e > en) e = en;
    float acc = 0.f;
    for (int n = b; n < e; ++n) acc += hsrc[(long)n * ld + c];
    if (b < e) atomicAdd(&vt[(g << 7) + c], acc);
}

// ---------------- vn = relu(relu((vt+vn)@W1+b1)@W2+b2); vt = 0 ---------------
__global__ void vn_update(float* __restrict__ vt, const float* __restrict__ W1,
                          const float* __restrict__ b1, const float* __restrict__ W2,
                          const float* __restrict__ b2, float* __restrict__ vn)
{
    __shared__ float row[HD];
    __shared__ float u[HD];
    int g = blockIdx.x, c = threadIdx.x;
    row[c] = vt[(g << 7) + c] + vn[(g << 7) + c];
    __syncthreads();
    float s = b1[c];
    for (int k = 0; k < HD; ++k) s += row[k] * W1[k * HD + c];
    u[c] = fmaxf(s, 0.f);
    __syncthreads();
    float s2 = b2[c];
    for (int k = 0; k < HD; ++k) s2 += u[k] * W2[k * HD + c];
    vn[(g << 7) + c] = fmaxf(s2, 0.f);
    vt[(g << 7) + c] = 0.f;                   // ready for next layer
}

// =============================================================================
extern "C" void kernel_launch(void* const* d_in, const int* in_sizes, int n_in,
                              void* d_out, int out_size, void* d_ws, size_t ws_size,
                              hipStream_t stream)
{
    const float* x         = (const float*)d_in[0];
    const float* edge_attr = (const float*)d_in[1];
    const float* conv_W1   = (const float*)d_in[2];
    const float* conv_b1   = (const float*)d_in[3];
    const float* conv_W2   = (const float*)d_in[4];
    const float* conv_b2   = (const float*)d_in[5];
    const float* conv_eps  = (const float*)d_in[6];
    const float* edge_W    = (const float*)d_in[7];
    const float* edge_b    = (const float*)d_in[8];
    const float* vn_W1     = (const float*)d_in[9];
    const float* vn_b1     = (const float*)d_in[10];
    const float* vn_W2     = (const float*)d_in[11];
    const float* vn_b2     = (const float*)d_in[12];
    const float* vn_emb    = (const float*)d_in[13];
    const int*   edge_index= (const int*)d_in[14];
    const int*   batch     = (const int*)d_in[15];
    float* out = (float*)d_out;

    const int L = in_sizes[6];                // conv_eps has L entries
    const int G = 64;
    const int N = in_sizes[0] / HD;
    const int E = in_sizes[1];
    const long NH = (long)N * HD;
    const long ldout = (long)L * HD;

    // workspace carve (h, z, t/agg reuse one buffer each; all fp32)
    char* p = (char*)d_ws;
    auto carve = [&](size_t bytes) -> char* {
        char* r = p; p += (bytes + 255) & ~(size_t)255; return r;
    };
    float* hbuf  = (float*)carve(NH * 4);
    float* zbuf  = (float*)carve(NH * 4);
    float* tbuf  = (float*)carve(NH * 4);     // doubles as agg then t
    float* deg   = (float*)carve((size_t)N * 4);
    float* vn    = (float*)carve((size_t)G * HD * 4);
    float* vt    = (float*)carve((size_t)G * HD * 4);
    int*   starts= (int*)  carve((size_t)(G + 1) * 4);

    const int* src = edge_index;
    const int* tgt = edge_index + E;

    init_all<<<(N + 255) / 256, 256, 0, stream>>>(deg, vn, vt, vn_emb, batch, starts, N, G);
    deg_accum<<<(E + 255) / 256, 256, 0, stream>>>(tgt, deg, E);

    const int gNH   = (int)((NH + 255) / 256);
    const int gGemm = (N + 127) / 128;
    for (int i = 0; i < L; ++i) {
        const float* hin = (i == 0) ? x : out + (long)(i - 1) * HD;
        long ldin        = (i == 0) ? (long)HD : ldout;

        add_vn_zero_agg<<<gNH, 256, 0, stream>>>(hin, ldin, vn, batch, hbuf, tbuf, N);
        edge_scatter<<<E, HD, 0, stream>>>(hbuf, edge_attr, src, tgt,
                                           edge_W + (long)i * HD, edge_b + (long)i * HD, tbuf);
        make_z<<<gNH, 256, 0, stream>>>(hbuf, tbuf, deg, conv_eps, i, zbuf, N);

        gemm_wmma_f32<<<gGemm, 256, 0, stream>>>(zbuf, conv_W1 + (long)i * HD * HD,
                                                 conv_b1 + (long)i * HD, tbuf, HD, N, 1);
        gemm_wmma_f32<<<gGemm, 256, 0, stream>>>(tbuf, conv_W2 + (long)i * HD * HD,
                                                 conv_b2 + (long)i * HD,
                                                 out + (long)i * HD, ldout, N, 0);

        if (i < L - 1) {
            vt_accum<<<dim3(G, 16), HD, 0, stream>>>(out + (long)i * HD, ldout, starts, vt);
            vn_update<<<G, HD, 0, stream>>>(vt, vn_W1 + (long)i * HD * HD, vn_b1 + (long)i * HD,
                                            vn_W2 + (long)i * HD * HD, vn_b2 + (long)i * HD, vn);
        }
    }
}